// DiscriminatorAD_6055903887557
// MI455X (gfx1250) — compile-verified
//
#include <hip/hip_runtime.h>
#include <math.h>

// CDNA5 / gfx1250: wave32, WMMA f32 16x16x4.
typedef __attribute__((ext_vector_type(2))) float v2f;
typedef __attribute__((ext_vector_type(4))) float v4f;
typedef __attribute__((ext_vector_type(8))) float v8f;

// ---------------------------------------------------------------------------
// Kernel 1: build support1 = x @ W1 directly in WMMA B-fragment layout.
// For each K-block k0 (step 4): 32 lanes x 2 floats contiguous (256 B):
//   s1B[(k0/4)*64 + lane*2 + v] = support1[k0 + 2*(lane>>4) + v][lane&15]
// (cols 5..15 zero). B-fragment (4x16 f32): lanes 0-15 hold K={0,1} at
// N=lane, lanes 16-31 hold K={2,3} at N=lane-16; V0=first K, V1=second.
// ---------------------------------------------------------------------------
__global__ void k_build_s1B(const float* __restrict__ x,
                            const float* __restrict__ W1,
                            float* __restrict__ s1B, int N) {
  int idx = blockIdx.x * blockDim.x + threadIdx.x;
  int total = (N >> 2) * 64;
  if (idx >= total) return;
  int v    = idx & 1;
  int lane = (idx >> 1) & 31;
  int blk  = idx >> 6;
  int k = blk * 4 + 2 * (lane >> 4) + v;
  int c = lane & 15;
  float val = 0.0f;
  if (c < 5 && k < N) {
    // W1 is [2][5] row-major
    val = x[2 * k] * W1[c] + x[2 * k + 1] * W1[5 + c];
  }
  s1B[idx] = val;  // idx == blk*64 + lane*2 + v by construction
}

// ---------------------------------------------------------------------------
// Kernel 2: pass 1 — hpart[chunk][row][0..4] partial of adj @ support1.
// One wave per (16-row tile, K-chunk). Wave id is forced into an SGPR via
// readfirstlane so the K loop is a *scalar* loop: full EXEC for WMMA, no
// per-iteration mask bookkeeping, pointer-increment address streams, and
// unroll-4 keeps 8 loads in flight instead of s_wait_loadcnt 0 per wmma.
// A-fragment (16x4 f32): lane L reads adj[row0+(L&15)][k + 2*(L>>4) + {0,1}]
// as one float2; each 128B adj line is reused over 8 consecutive K-steps in
// L0, so these loads are regular-temporal (NOT non-temporal).
// ---------------------------------------------------------------------------
__global__ void k_pass1(const float* __restrict__ adj,
                        const float* __restrict__ s1B,
                        float* __restrict__ hpart, int N, int nch) {
  int wslot = __builtin_amdgcn_readfirstlane(threadIdx.x >> 5);   // SGPR
  int wid   = blockIdx.x * (int)(blockDim.x >> 5) + wslot;        // SGPR
  int lane  = threadIdx.x & 31;
  int ntiles = N >> 4;
  if (wid >= ntiles * nch) return;  // scalar branch; EXEC stays all-1

  int tile  = wid / nch;
  int chunk = wid - tile * nch;
  int row0  = tile << 4;

  int nblk   = N >> 2;                                   // K-blocks of 4
  int bstart = (int)(((long long)chunk)     * nblk / nch);
  int bend   = (int)(((long long)chunk + 1) * nblk / nch);

  int lm = lane & 15, lh = lane >> 4;
  const float* ap = adj + (size_t)(row0 + lm) * (size_t)N + 2 * lh
                        + ((size_t)bstart << 2);
  const float* bp = s1B + (size_t)lane * 2 + (size_t)bstart * 64;

  v8f acc = {};
#pragma unroll 4
  for (int b = bstart; b < bend; ++b) {
    v2f a  = *(const v2f*)ap;   // adj fragment: L0-resident across 8 steps
    v2f bf = *(const v2f*)bp;   // s1B fragment: coalesced, L2-hot
    acc = __builtin_amdgcn_wmma_f32_16x16x4_f32(
        /*neg_a=*/false, a, /*neg_b=*/false, bf,
        /*c_mod=*/(short)0, acc, /*reuse_a=*/false, /*reuse_b=*/false);
    ap += 4;    // 16 B / step
    bp += 64;   // 256 B / step
  }

  // D layout: VGPR v, lanes 0-15 -> M=v, lanes 16-31 -> M=v+8; col = lane&15.
  int col = lm;
  if (col < 5) {
    size_t rbase = (size_t)(row0 + 8 * lh);
    float* dst = hpart + ((size_t)chunk * (size_t)N) * 8;
#pragma unroll
    for (int v = 0; v < 8; ++v) {
      dst[(rbase + v) * 8 + col] = acc[v];
    }
  }
}

// ---------------------------------------------------------------------------
// Kernel 3: reduce chunks (fixed order => deterministic), add b1, relu,
// contract with W2: h2[i] = sum_c relu(sum_ch hpart[ch][i][c] + b1[c])*W2[c]
// ---------------------------------------------------------------------------
__global__ void k_h2(const float* __restrict__ hpart,
                     const float* __restrict__ b1,
                     const float* __restrict__ W2,
                     float* __restrict__ h2, int N, int nch) {
  int i = blockIdx.x * blockDim.x + threadIdx.x;
  if (i >= N) return;
  float acc[5] = {0.f, 0.f, 0.f, 0.f, 0.f};
  for (int ch = 0; ch < nch; ++ch) {
    const float* p = hpart + ((size_t)ch * (size_t)N + (size_t)i) * 8;
#pragma unroll
    for (int c = 0; c < 5; ++c) acc[c] += p[c];
  }
  float s = 0.f;
#pragma unroll
  for (int c = 0; c < 5; ++c) {
    float hv = acc[c] + b1[c];
    hv = hv > 0.f ? hv : 0.f;
    s += hv * W2[c];
  }
  h2[i] = s;
}

// ---------------------------------------------------------------------------
// Kernel 4: pass 2 — wave per row j: sc[j] = lin_W[j] * (adj[j,:].h2 + b2).
// Scalar wave id + uniform trip count (scalar loop) with one predicated
// remainder load. adj is use-once here -> non-temporal float4 streams;
// h2 (40 KB) stays cache-resident.
// ---------------------------------------------------------------------------
__global__ void k_pass2(const float* __restrict__ adj,
                        const float* __restrict__ h2,
                        const float* __restrict__ lin_W,
                        const float* __restrict__ b2,
                        float* __restrict__ sc, int N) {
  int wslot = __builtin_amdgcn_readfirstlane(threadIdx.x >> 5);   // SGPR
  int wid   = blockIdx.x * (int)(blockDim.x >> 5) + wslot;        // SGPR
  int lane  = threadIdx.x & 31;
  if (wid >= N) return;  // scalar branch

  const float* row = adj + (size_t)wid * (size_t)N;
  const float* ap  = row + lane * 4;
  const float* hp  = h2 + lane * 4;

  float sum = 0.f;
  int nfull = N >> 7;  // full 128-element sweeps (uniform trip count)
#pragma unroll 2
  for (int it = 0; it < nfull; ++it) {
    v4f a = __builtin_nontemporal_load((const v4f*)ap);
    v4f h = *(const v4f*)hp;
    sum += a.x * h.x + a.y * h.y + a.z * h.z + a.w * h.w;
    ap += 128;
    hp += 128;
  }
  int irem = (nfull << 7) + lane * 4;
  if (irem < N) {  // N % 4 == 0 -> always a full float4
    v4f a = __builtin_nontemporal_load((const v4f*)(row + irem));
    v4f h = *(const v4f*)(h2 + irem);
    sum += a.x * h.x + a.y * h.y + a.z * h.z + a.w * h.w;
  }

#pragma unroll
  for (int off = 16; off > 0; off >>= 1) sum += __shfl_xor(sum, off, 32);
  if (lane == 0) sc[wid] = lin_W[wid] * (sum + b2[0]);
}

// ---------------------------------------------------------------------------
// Kernel 5: logits = sum_j sc[j] + lin_b; out = sigmoid(logits)
// ---------------------------------------------------------------------------
__global__ void k_final(const float* __restrict__ sc,
                        const float* __restrict__ lin_b,
                        float* __restrict__ out, int N) {
  __shared__ float red[32];
  int tid = threadIdx.x;
  float s = 0.f;
  for (int i = tid; i < N; i += blockDim.x) s += sc[i];
#pragma unroll
  for (int off = 16; off > 0; off >>= 1) s += __shfl_xor(s, off, 32);
  int wav = tid >> 5, lane = tid & 31;
  if (lane == 0) red[wav] = s;
  __syncthreads();
  if (wav == 0) {
    int nw = (int)(blockDim.x >> 5);
    float t = (lane < nw) ? red[lane] : 0.f;
#pragma unroll
    for (int off = 16; off > 0; off >>= 1) t += __shfl_xor(t, off, 32);
    if (lane == 0) {
      float logits = t + lin_b[0];
      out[0] = 1.f / (1.f + expf(-logits));
    }
  }
}

// ---------------------------------------------------------------------------
// Launch
// ---------------------------------------------------------------------------
extern "C" void kernel_launch(void* const* d_in, const int* in_sizes, int n_in,
                              void* d_out, int out_size, void* d_ws, size_t ws_size,
                              hipStream_t stream) {
  const float* x     = (const float*)d_in[0];
  const float* adj   = (const float*)d_in[1];
  const float* W1    = (const float*)d_in[2];
  const float* b1    = (const float*)d_in[3];
  const float* W2    = (const float*)d_in[4];
  const float* b2    = (const float*)d_in[5];
  const float* lin_W = (const float*)d_in[6];
  const float* lin_b = (const float*)d_in[7];
  float* out = (float*)d_out;

  const int N = in_sizes[0] / 2;  // x is [N,2] -> N = 10000 (N%16==0, N%4==0)

  // K-chunk count for pass 1 (parallelism for latency hiding); shrink if ws
  // is tight. ws need: (16 + 8*nch + 2) * N floats. Deterministic per run.
  int nch = 10;
  {
    size_t perN = ws_size / ((size_t)N * sizeof(float));
    if (perN < 18u + 8u * (size_t)nch) {
      long long avail = (long long)perN - 18;
      nch = avail >= 8 ? (int)(avail / 8) : 1;
      if (nch < 1) nch = 1;
      if (nch > 10) nch = 10;
    }
  }

  float* s1B   = (float*)d_ws;                        // 16*N floats
  float* hpart = s1B + (size_t)16 * N;                // 8*N*nch floats
  float* h2    = hpart + (size_t)8 * N * (size_t)nch; // N floats
  float* sc    = h2 + N;                              // N floats

  const int B = 256;  // 8 waves / block

  int n16 = 16 * N;
  k_build_s1B<<<(n16 + B - 1) / B, B, 0, stream>>>(x, W1, s1B, N);

  int waves1 = (N >> 4) * nch;
  k_pass1<<<(waves1 * 32 + B - 1) / B, B, 0, stream>>>(adj, s1B, hpart, N, nch);

  k_h2<<<(N + B - 1) / B, B, 0, stream>>>(hpart, b1, W2, h2, N, nch);

  k_pass2<<<((size_t)N * 32 + B - 1) / B, B, 0, stream>>>(adj, h2, lin_W, b2, sc, N);

  k_final<<<1, 1024, 0, stream>>>(sc, lin_b, out, N);
}